// Sparsemax_54339926229490
// MI455X (gfx1250) — compile-verified
//
#include <hip/hip_runtime.h>

// Sparsemax along the contiguous axis (K = 2048) of an (8, 1024, 2048) f32 tensor.
// Bandwidth-bound: 128 MiB total HBM traffic -> ~5.6 us floor at 23.3 TB/s.
// One workgroup (256 threads = 8 wave32 waves) per row. Row is staged into LDS
// via CDNA5 async global->LDS b128 copies (ASYNCcnt), read once from LDS into registers.

#define KDIM 2048
#define THREADS 256
#define VT 8  // KDIM / THREADS elements per thread

__global__ __launch_bounds__(THREADS)
void sparsemax_row_kernel(const float* __restrict__ x, float* __restrict__ out) {
    __shared__ __align__(16) float srow[KDIM];
    __shared__ float swsum[8];
    __shared__ float swk[8];
    __shared__ float swt[8];

    const int tid  = threadIdx.x;
    const int lane = tid & 31;
    const int wid  = tid >> 5;
    const size_t rowBase = (size_t)blockIdx.x * KDIM;
    const float* gsrc = x + rowBase;

    // ---- Stage row into LDS with async global->LDS copies (CDNA5 ASYNCcnt path).
    // 256 lanes x 16B x 2 rounds = 8192 B = one full row.
    {
        const float* g0 = gsrc + tid * 4;
        const float* g1 = gsrc + 1024 + tid * 4;
        unsigned l0 = (unsigned)(unsigned long long)(void*)&srow[tid * 4];
        unsigned l1 = (unsigned)(unsigned long long)(void*)&srow[1024 + tid * 4];
        asm volatile("global_load_async_to_lds_b128 %0, %1, off"
                     :: "v"(l0), "v"(g0) : "memory");
        asm volatile("global_load_async_to_lds_b128 %0, %1, off"
                     :: "v"(l1), "v"(g1) : "memory");
        asm volatile("s_wait_asynccnt 0x0" ::: "memory");
    }
    __syncthreads();

    // ---- Per-thread segment: local inclusive cumsum.
    float z[VT];
    {
        const float4* sv = (const float4*)&srow[tid * VT];
        float4 a = sv[0], b = sv[1];
        z[0] = a.x; z[1] = a.y; z[2] = a.z; z[3] = a.w;
        z[4] = b.x; z[5] = b.y; z[6] = b.z; z[7] = b.w;
    }
    float c[VT];
    float run = 0.0f;
#pragma unroll
    for (int i = 0; i < VT; ++i) { run += z[i]; c[i] = run; }

    // ---- Wave32 inclusive scan of per-thread sums.
    float incl = run;
#pragma unroll
    for (int d = 1; d < 32; d <<= 1) {
        float n = __shfl_up(incl, (unsigned)d);
        if (lane >= d) incl += n;
    }
    if (lane == 31) swsum[wid] = incl;
    __syncthreads();

    // Cross-wave exclusive prefix (8 entries: cheap serial accumulate).
    float wexcl = 0.0f;
#pragma unroll
    for (int w = 0; w < 8; ++w) {
        float s = swsum[w];
        if (w < wid) wexcl += s;
    }
    const float texcl = wexcl + (incl - run);  // exclusive prefix of this thread's segment

    // ---- Mask test and local maxima of k (as float) and of masked cumsum.
    float kmax = 0.0f;
    float tmpmax = -__builtin_inff();
    const int gbase = tid * VT;
#pragma unroll
    for (int i = 0; i < VT; ++i) {
        float cc   = texcl + c[i];
        float kk   = (float)(gbase + i + 1);
        float left = 1.0f + kk * z[i];
        if (!(left <= cc)) {           // mask == False positions
            kmax   = fmaxf(kmax, kk);
            tmpmax = fmaxf(tmpmax, cc);
        }
    }

    // ---- Wave max-reduce, then cross-wave via LDS.
#pragma unroll
    for (int d = 16; d > 0; d >>= 1) {
        kmax   = fmaxf(kmax,   __shfl_xor(kmax,   d));
        tmpmax = fmaxf(tmpmax, __shfl_xor(tmpmax, d));
    }
    if (lane == 0) { swk[wid] = kmax; swt[wid] = tmpmax; }
    __syncthreads();

    float kfin = swk[0], tfin = swt[0];
#pragma unroll
    for (int w = 1; w < 8; ++w) {
        kfin = fmaxf(kfin, swk[w]);
        tfin = fmaxf(tfin, swt[w]);
    }

    const float tau = (tfin - 1.0f) / kfin;

    // ---- Output: relu(z - tau), two coalesced b128 stores per thread.
    float4 o0, o1;
    o0.x = fmaxf(z[0] - tau, 0.0f);
    o0.y = fmaxf(z[1] - tau, 0.0f);
    o0.z = fmaxf(z[2] - tau, 0.0f);
    o0.w = fmaxf(z[3] - tau, 0.0f);
    o1.x = fmaxf(z[4] - tau, 0.0f);
    o1.y = fmaxf(z[5] - tau, 0.0f);
    o1.z = fmaxf(z[6] - tau, 0.0f);
    o1.w = fmaxf(z[7] - tau, 0.0f);
    float4* op = (float4*)(out + rowBase + (size_t)tid * VT);
    op[0] = o0;
    op[1] = o1;
}

extern "C" void kernel_launch(void* const* d_in, const int* in_sizes, int n_in,
                              void* d_out, int out_size, void* d_ws, size_t ws_size,
                              hipStream_t stream) {
    const float* x = (const float*)d_in[0];
    float* out = (float*)d_out;
    const int rows = in_sizes[0] / KDIM;  // 8 * 1024 = 8192 rows of 2048
    sparsemax_row_kernel<<<rows, THREADS, 0, stream>>>(x, out);
}